// GraphNeuralNetwork_50964081934784
// MI455X (gfx1250) — compile-verified
//
#include <hip/hip_runtime.h>

// GCN: 3 x (dense GEMM via V_WMMA_F32_16X16X4_F32 + bias-seeded COO scatter-add)
// MI455X reasoning: SpMM phase is L2-bound (51MB working set << 192MB L2),
// GEMM is only ~4.1 GFLOP -> use exact fp32 WMMA, fuse ReLU into GEMM input staging.

typedef __attribute__((ext_vector_type(2))) float v2f;
typedef __attribute__((ext_vector_type(8))) float v8f;

#define NNODES 50000
#define NEDGES 800000
#define DIN    128
#define DHID   128
#define DOUT   64
#define LDS_STRIDE 132   // 132 % 64 == 4 -> conflict-free A-fragment ds_loads

// ---------------------------------------------------------------------------
// GEMM: S[rowBase..rowBase+15, 0..dout) = relu?(X[rowBase.., 0..128)) @ W
// One block per 16-row tile. Wave w computes the 16x16 column tile at w*16.
// A-fragment layout (32-bit A 16x4, ISA 7.12.2): lanes 0-15 M=lane hold K=0,1
// in v0,v1; lanes 16-31 hold K=2,3.  C/D: VGPR r = row (r + 8*half), col=lidx.
// ---------------------------------------------------------------------------
template <bool RELU_IN>
__global__ __launch_bounds__(256) void gcn_gemm(const float* __restrict__ X,
                                                const float* __restrict__ W,
                                                float* __restrict__ S,
                                                int dout) {
  __shared__ float xs[16 * LDS_STRIDE];

  const int rowBase = blockIdx.x * 16;
  const int tid     = threadIdx.x;
  const int nthr    = blockDim.x;

  // Stage the contiguous 16x128 input tile into LDS (ReLU fused for layers 2/3).
  const float4* src = (const float4*)(X + (size_t)rowBase * DIN);
  for (int i = tid; i < 16 * DIN / 4; i += nthr) {
    float4 v = src[i];
    if (RELU_IN) {
      v.x = fmaxf(v.x, 0.f); v.y = fmaxf(v.y, 0.f);
      v.z = fmaxf(v.z, 0.f); v.w = fmaxf(v.w, 0.f);
    }
    const int r = (i * 4) / DIN;
    const int c = (i * 4) % DIN;
    *(float4*)&xs[r * LDS_STRIDE + c] = v;
  }
  __syncthreads();

  const int wave    = tid >> 5;
  const int lane    = tid & 31;
  const int half    = lane >> 4;   // 0: K=0,1  1: K=2,3
  const int lidx    = lane & 15;
  const int colBase = wave * 16;

  v8f acc = {};
  for (int k0 = 0; k0 < DIN; k0 += 4) {
    const int ka = k0 + 2 * half;
    v2f a, b;
    a.x = xs[lidx * LDS_STRIDE + ka];
    a.y = xs[lidx * LDS_STRIDE + ka + 1];
    b.x = W[(size_t)ka * dout + colBase + lidx];
    b.y = W[(size_t)(ka + 1) * dout + colBase + lidx];
    acc = __builtin_amdgcn_wmma_f32_16x16x4_f32(
        /*neg_a=*/false, a, /*neg_b=*/false, b,
        /*c_mod=*/(short)0, acc, /*reuse_a=*/false, /*reuse_b=*/false);
  }

  float* dst = S + (size_t)rowBase * dout + colBase + lidx;
#pragma unroll
  for (int r = 0; r < 8; ++r) {
    dst[(size_t)(r + 8 * half) * dout] = acc[r];
  }
}

// ---------------------------------------------------------------------------
// agg[i][j] = bias[j]   (bias folded in before the scatter-add)
// ---------------------------------------------------------------------------
__global__ __launch_bounds__(256) void gcn_init_bias(float* __restrict__ agg,
                                                     const float* __restrict__ bias,
                                                     int n, int doutMask) {
  const int i = blockIdx.x * 256 + threadIdx.x;
  if (i < n) agg[i] = bias[i & doutMask];
}

// ---------------------------------------------------------------------------
// One wave per edge: agg[row] += val * S[col]   (global_atomic_add_f32)
// VPL = dout/32 floats per lane -> coalesced 512B/256B row gather.
// ---------------------------------------------------------------------------
template <int VPL>
__global__ __launch_bounds__(256) void gcn_scatter(const float* __restrict__ S,
                                                   const int* __restrict__ erow,
                                                   const int* __restrict__ ecol,
                                                   const float* __restrict__ eval,
                                                   float* __restrict__ agg) {
  const int e = blockIdx.x * 8 + (threadIdx.x >> 5);
  if (e >= NEDGES) return;
  const int lane = threadIdx.x & 31;
  const int row  = erow[e];
  const int col  = ecol[e];
  const float w  = eval[e];

  constexpr int dout = VPL * 32;
  const float* srow = S + (size_t)col * dout + lane * VPL;
  float* drow       = agg + (size_t)row * dout + lane * VPL;

  float v[VPL];
#pragma unroll
  for (int j = 0; j < VPL; ++j) v[j] = srow[j];   // merges into b128/b64 load
#pragma unroll
  for (int j = 0; j < VPL; ++j) atomicAdd(&drow[j], w * v[j]);
}

// ---------------------------------------------------------------------------
extern "C" void kernel_launch(void* const* d_in, const int* in_sizes, int n_in,
                              void* d_out, int out_size, void* d_ws, size_t ws_size,
                              hipStream_t stream) {
  (void)in_sizes; (void)n_in; (void)out_size; (void)ws_size;

  const float* x    = (const float*)d_in[0];
  const int*   erow = (const int*)d_in[1];
  const int*   ecol = (const int*)d_in[2];
  const float* eval = (const float*)d_in[3];
  const float* W1   = (const float*)d_in[4];
  const float* b1   = (const float*)d_in[5];
  const float* W2   = (const float*)d_in[6];
  const float* b2   = (const float*)d_in[7];
  const float* W3   = (const float*)d_in[8];
  const float* b3   = (const float*)d_in[9];
  float* out = (float*)d_out;

  float* bufS = (float*)d_ws;                      // support  [50000 x 128]
  float* bufA = bufS + (size_t)NNODES * DHID;      // agg      [50000 x 128]

  const int rowTiles      = NNODES / 16;   // 3125 (exact)
  const int scatterBlocks = NEDGES / 8;    // 100000 (exact), 8 waves/block
  const int nHid = NNODES * DHID;
  const int nOut = NNODES * DOUT;

  // ---- Layer 1: h1 = relu(b1 + A @ (x @ W1)); relu deferred to layer-2 GEMM read
  gcn_gemm<false><<<rowTiles, 256, 0, stream>>>(x, W1, bufS, DHID);
  gcn_init_bias<<<(nHid + 255) / 256, 256, 0, stream>>>(bufA, b1, nHid, DHID - 1);
  gcn_scatter<4><<<scatterBlocks, 256, 0, stream>>>(bufS, erow, ecol, eval, bufA);

  // ---- Layer 2
  gcn_gemm<true><<<rowTiles, 256, 0, stream>>>(bufA, W2, bufS, DHID);
  gcn_init_bias<<<(nHid + 255) / 256, 256, 0, stream>>>(bufA, b2, nHid, DHID - 1);
  gcn_scatter<4><<<scatterBlocks, 256, 0, stream>>>(bufS, erow, ecol, eval, bufA);

  // ---- Layer 3 (no final activation), scatter straight into d_out
  gcn_gemm<true><<<rowTiles, 128, 0, stream>>>(bufA, W3, bufS, DOUT);
  gcn_init_bias<<<(nOut + 255) / 256, 256, 0, stream>>>(out, b3, nOut, DOUT - 1);
  gcn_scatter<2><<<scatterBlocks, 256, 0, stream>>>(bufS, erow, ecol, eval, out);
}